// M3S_75127567942075
// MI455X (gfx1250) — compile-verified
//
#include <hip/hip_runtime.h>
#include <hip/hip_bf16.h>

typedef float v2f __attribute__((ext_vector_type(2)));
typedef float v8f __attribute__((ext_vector_type(8)));

#define N_NODES 100000
#define N_EDGES 1600000
#define F_IN    256
#define F_HID   128
#define F_CLS   40

// ---------------------------------------------------------------- utilities

__global__ void zero_i32(int* __restrict__ p, size_t n) {
  size_t i = (size_t)blockIdx.x * blockDim.x + threadIdx.x;
  size_t stride = (size_t)gridDim.x * blockDim.x;
  for (; i < n; i += stride) p[i] = 0;
}

__global__ void degree_accum(const int* __restrict__ src, const int* __restrict__ dst,
                             int* __restrict__ outdeg, int* __restrict__ indeg, int E) {
  int i = blockIdx.x * blockDim.x + threadIdx.x;
  if (i < E) {
    atomicAdd(&outdeg[src[i]], 1);
    atomicAdd(&indeg[dst[i]], 1);
  }
}

__global__ void finish_norms(const int* __restrict__ outdeg, const int* __restrict__ indeg,
                             float* __restrict__ cs, float* __restrict__ cd, int n) {
  int i = blockIdx.x * blockDim.x + threadIdx.x;
  if (i < n) {
    cs[i] = rsqrtf(fmaxf((float)outdeg[i], 1.0f));
    cd[i] = rsqrtf(fmaxf((float)indeg[i], 1.0f));
  }
}

// ---------------------------------------------------------------- CSR build
// Exclusive scan of in-degrees -> row offsets (block scan in LDS + block-sum fixup).

__global__ void scan_block(const int* __restrict__ in, int* __restrict__ out,
                           int* __restrict__ bsum, int n) {
  __shared__ int sh[256];
  int gid = blockIdx.x * 256 + threadIdx.x;
  int v = (gid < n) ? in[gid] : 0;
  sh[threadIdx.x] = v;
  __syncthreads();
  for (int off = 1; off < 256; off <<= 1) {
    int t = (threadIdx.x >= off) ? sh[threadIdx.x - off] : 0;
    __syncthreads();
    sh[threadIdx.x] += t;
    __syncthreads();
  }
  int incl = sh[threadIdx.x];
  if (gid < n) out[gid] = incl - v;                 // exclusive
  if (threadIdx.x == 255) bsum[blockIdx.x] = incl;  // block total
}

__global__ void scan_sums(int* __restrict__ bsum, int nb) {
  if (threadIdx.x == 0 && blockIdx.x == 0) {
    int running = 0;
    for (int i = 0; i < nb; ++i) {
      int t = bsum[i];
      bsum[i] = running;
      running += t;
    }
  }
}

__global__ void scan_add(int* __restrict__ out, const int* __restrict__ bsum,
                         int n, int total) {
  int gid = blockIdx.x * blockDim.x + threadIdx.x;
  if (gid < n) out[gid] += bsum[gid >> 8];
  if (gid == 0) out[n] = total;
}

__global__ void bin_edges(const int* __restrict__ src, const int* __restrict__ dst,
                          const int* __restrict__ row_off, int* __restrict__ cursor,
                          int* __restrict__ esrc, int E) {
  int e = blockIdx.x * blockDim.x + threadIdx.x;
  if (e < E) {
    int d = dst[e];
    int pos = atomicAdd(&cursor[d], 1);
    esrc[row_off[d] + pos] = src[e];
  }
}

// ---------------------------------------------------------------- WMMA GEMM
// out[M x NC] = A[M x K] @ W[K x NC], then optionally *rowscale[row] and +bias[col].
// One wave computes a 16-row x (NT*16)-col strip using V_WMMA_F32_16X16X4_F32.
template<int NT, int NC>
__global__ __launch_bounds__(256) void gemm_wmma(
    const float* __restrict__ A, const float* __restrict__ W,
    float* __restrict__ out,
    const float* __restrict__ rowscale,   // nullable
    const float* __restrict__ bias,       // nullable
    int M, int K)
{
  const int lane = threadIdx.x & 31;
  const int wave = threadIdx.x >> 5;
  const int rb = (blockIdx.x * 8 + wave) * 16;     // 8 waves per block
  if (rb >= M) return;                             // wave-uniform exit

  const int hi = lane >> 4;
  const int lo = lane & 15;

  v8f acc[NT] = {};
  const float* ap = A + (size_t)(rb + lo) * K + 2 * hi;

  for (int k = 0; k < K; k += 4) {
    v2f a = *(const v2f*)ap;
    ap += 4;
    const float* wrow = W + (size_t)(k + 2 * hi) * NC;
#pragma unroll
    for (int t = 0; t < NT; ++t) {
      int col = t * 16 + lo;
      v2f b;
      if (col < NC) { b.x = wrow[col]; b.y = wrow[col + NC]; }
      else          { b.x = 0.0f;      b.y = 0.0f;           }
      acc[t] = __builtin_amdgcn_wmma_f32_16x16x4_f32(
          false, a, false, b, (short)0, acc[t], false, false);
    }
  }

  float sc[8];
#pragma unroll
  for (int r = 0; r < 8; ++r)
    sc[r] = rowscale ? rowscale[rb + hi * 8 + r] : 1.0f;

#pragma unroll
  for (int t = 0; t < NT; ++t) {
    int col = t * 16 + lo;
    if (col < NC) {
      float bb = bias ? bias[col] : 0.0f;
#pragma unroll
      for (int r = 0; r < 8; ++r) {
        out[(size_t)(rb + hi * 8 + r) * NC + col] = acc[t][r] * sc[r] + bb;
      }
    }
  }
}

// ---------------------------------------------------------------- gather aggregation
// One wave per destination node: sum rows hs[esrc[j]] over its CSR segment,
// then fused epilogue  h = sum * c_dst[n] + bias;  outRelu = relu(h); optional raw h.
// Edge indices are loaded 32-at-a-time and broadcast lane-to-lane via __shfl
// (ds_bpermute on wave32); feature loads are one float4 per lane (512B/row, coalesced).
__global__ __launch_bounds__(256) void gather_conv(
    const float* __restrict__ hs, const int* __restrict__ esrc,
    const int* __restrict__ row_off, const float* __restrict__ cdst,
    const float* __restrict__ bias, float* __restrict__ outRelu,
    float* __restrict__ outRaw, int n)
{
  int wid = (int)(((size_t)blockIdx.x * blockDim.x + threadIdx.x) >> 5);
  int lane = threadIdx.x & 31;
  if (wid >= n) return;

  int start = row_off[wid];
  int end = row_off[wid + 1];

  float4 acc = {0.0f, 0.0f, 0.0f, 0.0f};
  for (int base = start; base < end; base += 32) {
    int idx = base + lane;
    int myidx = (idx < end) ? esrc[idx] : 0;
    int m = end - base;
    if (m > 32) m = 32;
    for (int j = 0; j < m; ++j) {
      int s = __shfl(myidx, j);
      float4 v = ((const float4*)(hs + (size_t)s * F_HID))[lane];
      acc.x += v.x; acc.y += v.y; acc.z += v.z; acc.w += v.w;
    }
  }

  float c = cdst[wid];
  float4 b = ((const float4*)bias)[lane];
  float4 h;
  h.x = acc.x * c + b.x;
  h.y = acc.y * c + b.y;
  h.z = acc.z * c + b.z;
  h.w = acc.w * c + b.w;
  if (outRaw) ((float4*)outRaw)[(size_t)wid * 32 + lane] = h;
  float4 r;
  r.x = fmaxf(h.x, 0.0f);
  r.y = fmaxf(h.y, 0.0f);
  r.z = fmaxf(h.z, 0.0f);
  r.w = fmaxf(h.w, 0.0f);
  ((float4*)outRelu)[(size_t)wid * 32 + lane] = r;
}

// ---------------------------------------------------------------- launch

extern "C" void kernel_launch(void* const* d_in, const int* in_sizes, int n_in,
                              void* d_out, int out_size, void* d_ws, size_t ws_size,
                              hipStream_t stream) {
  const float* x   = (const float*)d_in[0];
  const int*   src = (const int*)  d_in[1];
  const int*   dst = (const int*)  d_in[2];
  const float* W0  = (const float*)d_in[3];
  const float* b0  = (const float*)d_in[4];
  const float* W1  = (const float*)d_in[5];
  const float* b1  = (const float*)d_in[6];
  const float* Wc  = (const float*)d_in[7];
  const float* bc  = (const float*)d_in[8];

  float* out_h  = (float*)d_out;                         // N x 128
  float* out_lg = out_h + (size_t)N_NODES * F_HID;       // N x 40

  float* cs     = (float*)d_ws;                          // N  (c_src)
  float* cd     = cs + N_NODES;                          // N  (c_dst)
  float* bufA   = cd + N_NODES;                          // N x 128 (relu activations)
  float* bufH   = bufA + (size_t)N_NODES * F_HID;        // N x 128 (gemm outputs)
  int*   indeg  = (int*)(bufH + (size_t)N_NODES * F_HID);// N
  int*   outdeg = indeg + N_NODES;                       // N
  int*   cursor = outdeg + N_NODES;                      // N
  int*   row_off= cursor + N_NODES;                      // N+1
  int*   bsums  = row_off + N_NODES + 2;                 // ~391 (padded)
  int*   esrc   = bsums + 512;                           // E

  const int TB = 256;
  const int nb_scan     = (N_NODES + TB - 1) / TB;       // 391
  const int gemm_blocks = ((N_NODES + 15) / 16 + 7) / 8; // 782
  const int node_blocks = (N_NODES + TB - 1) / TB;
  const int edge_blocks = (N_EDGES + TB - 1) / TB;
  const int wave_blocks = (N_NODES * 32 + TB - 1) / TB;  // 1 wave per node

  // --- CSR + degree norms (rebuilt every call: deterministic, replay-safe) ---
  zero_i32<<<2048, TB, 0, stream>>>(indeg, (size_t)3 * N_NODES);  // indeg,outdeg,cursor
  degree_accum<<<edge_blocks, TB, 0, stream>>>(src, dst, outdeg, indeg, N_EDGES);
  finish_norms<<<node_blocks, TB, 0, stream>>>(outdeg, indeg, cs, cd, N_NODES);
  scan_block<<<nb_scan, TB, 0, stream>>>(indeg, row_off, bsums, N_NODES);
  scan_sums<<<1, 32, 0, stream>>>(bsums, nb_scan);
  scan_add<<<node_blocks, TB, 0, stream>>>(row_off, bsums, N_NODES, N_EDGES);
  bin_edges<<<edge_blocks, TB, 0, stream>>>(src, dst, row_off, cursor, esrc, N_EDGES);

  // --- layer 0: bufA = relu( gather((x@W0)*c_src) * c_dst + b0 ) ---
  gemm_wmma<8, 128><<<gemm_blocks, TB, 0, stream>>>(x, W0, bufH, cs, nullptr, N_NODES, F_IN);
  gather_conv<<<wave_blocks, TB, 0, stream>>>(bufH, esrc, row_off, cd, b0, bufA, nullptr, N_NODES);

  // --- layer 1: h2 -> out_h (raw), bufA = relu(h2) ---
  gemm_wmma<8, 128><<<gemm_blocks, TB, 0, stream>>>(bufA, W1, bufH, cs, nullptr, N_NODES, F_HID);
  gather_conv<<<wave_blocks, TB, 0, stream>>>(bufH, esrc, row_off, cd, b1, bufA, out_h, N_NODES);

  // --- classifier: logits = relu(h2) @ Wc + bc ---
  gemm_wmma<3, 40><<<gemm_blocks, TB, 0, stream>>>(bufA, Wc, out_lg, nullptr, bc, N_NODES, F_HID);
}